// PointTransformer_51651276702286
// MI455X (gfx1250) — compile-verified
//
#include <hip/hip_runtime.h>
#include <math.h>

typedef __attribute__((ext_vector_type(2))) float v2f;
typedef __attribute__((ext_vector_type(4))) float v4f;
typedef __attribute__((ext_vector_type(8))) float v8f;

#define BATCH 2
#define NPTS  16384
#define KNBR  16
#define DIM   64

// ---- LDS layout (float offsets) ----
// 7 dense 64x64 weights, stored B-fragment swizzled:
//   Wswz[(row/4)*256 + col*4 + (row%4)] = W[row][col]
// -> fragment (rows 4ks+2hi, 4ks+2hi+1 @ col) is ONE aligned b64 at
//    ks*256 + col*4 + hi*2, and the two wave halves hit disjoint LDS banks.
#define W_OFF(i)    ((i) * 4096)
#define PW1_OFF     (7 * 4096)            // 64 cols x 4 rows: pW1 rows 0..2, row 3 = pb1
#define BPACK_OFF   (PW1_OFF + 256)       // 7 bias packs: 64 cols x 4 rows (row 0 = bias)
#define QTILE_OFF   (BPACK_OFF + 7 * 256) // 4 waves x 16x64
#define STAGE_STRIDE 68                   // padded row stride (conflict-free A reload)
#define STAGE_OFF   (QTILE_OFF + 4 * 1024)
#define SMEM_FLOATS (STAGE_OFF + 4 * (16 * STAGE_STRIDE))

#define UNROLL _Pragma("unroll")

__device__ __forceinline__ float swap16(float v) {
    // SWAPX16: group-of-32 swizzle, xor_mask=0x10, and_mask=0x1f
    return __int_as_float(__builtin_amdgcn_ds_swizzle(__float_as_int(v), 0x401F));
}

__device__ __forceinline__ void load_afrag_global(const float* base, v2f* af, int lo, int hi) {
    UNROLL for (int ks = 0; ks < 16; ++ks)
        af[ks] = *(const v2f*)(base + lo * DIM + ks * 4 + hi * 2);
}

__device__ __forceinline__ void load_afrag_stage(const float* stg, v2f* af, int lo, int hi) {
    UNROLL for (int ks = 0; ks < 16; ++ks)
        af[ks] = *(const v2f*)(stg + lo * STAGE_STRIDE + ks * 4 + hi * 2);
}

__device__ __forceinline__ void store_c_stage(float* stg, const v8f* c, int lo, int hi) {
    UNROLL for (int nb = 0; nb < 4; ++nb)
        UNROLL for (int r = 0; r < 8; ++r)
            stg[(r + 8 * hi) * STAGE_STRIDE + nb * 16 + lo] = c[nb][r];
}

// 16x64 = (16x64) @ (64x64) + bias, optional ReLU. f32 WMMA 16x16x4.
// Bias is applied by a leading WMMA k-step: A = (1,0)/(0,0) constant fragment,
// B = bias pack row 0, C = inline zero. Then 16 chained k-steps per N-tile,
// 4 independent accumulator chains interleaved (hazard-friendly).
__device__ __forceinline__ void mm16(const v2f* af, const float* Wswz, const float* bpack,
                                     v8f* cout, bool relu, int lo, int hi, v2f ab) {
    const v8f zacc = {0.f, 0.f, 0.f, 0.f, 0.f, 0.f, 0.f, 0.f};
    UNROLL for (int nb = 0; nb < 4; ++nb) {
        v2f bf = *(const v2f*)(bpack + (nb * 16 + lo) * 4 + hi * 2);
        cout[nb] = __builtin_amdgcn_wmma_f32_16x16x4_f32(
            false, ab, false, bf, (short)0, zacc, false, false);
    }
    UNROLL for (int ks = 0; ks < 16; ++ks) {
        const float* wp = Wswz + ks * 256 + hi * 2;
        UNROLL for (int nb = 0; nb < 4; ++nb) {
            v2f bf = *(const v2f*)(wp + (nb * 16 + lo) * 4);
            cout[nb] = __builtin_amdgcn_wmma_f32_16x16x4_f32(
                false, af[ks], false, bf, (short)0, cout[nb], false, false);
        }
    }
    if (relu) {
        UNROLL for (int nb = 0; nb < 4; ++nb)
            UNROLL for (int r = 0; r < 8; ++r)
                cout[nb][r] = fmaxf(cout[nb][r], 0.0f);
    }
}

__global__ __launch_bounds__(128) void pt_fused_kernel(
    const float* __restrict__ x,   const float* __restrict__ xn,
    const float* __restrict__ p,   const float* __restrict__ pn,
    const float* __restrict__ qW1, const float* __restrict__ qb1,
    const float* __restrict__ qW2, const float* __restrict__ qb2,
    const float* __restrict__ kW1, const float* __restrict__ kb1,
    const float* __restrict__ kW2, const float* __restrict__ kb2,
    const float* __restrict__ vW1, const float* __restrict__ vb1,
    const float* __restrict__ vW2, const float* __restrict__ vb2,
    const float* __restrict__ pW1, const float* __restrict__ pb1,
    const float* __restrict__ pW2, const float* __restrict__ pb2,
    float* __restrict__ out)
{
    extern __shared__ float smem[];
    const int tid  = threadIdx.x;
    const int wave = tid >> 5;
    const int lane = tid & 31;
    const int lo   = lane & 15;
    const int hi   = lane >> 4;

    // ---- cooperative weight staging into LDS (swizzled to fragment layout) ----
    {
        const float* wsrc[7] = {qW1, qW2, kW1, kW2, vW1, vW2, pW2};
        UNROLL for (int w = 0; w < 7; ++w) {
            const v4f* src = (const v4f*)wsrc[w];
            float* dst = &smem[W_OFF(w)];
            for (int i = tid; i < 1024; i += 128) {       // b128 coalesced reads
                v4f vv = src[i];
                int row = i >> 4;
                int colBase = (i & 15) * 4;
                int pr = row >> 2, rr = row & 3;
                UNROLL for (int e = 0; e < 4; ++e)
                    dst[(pr * 64 + colBase + e) * 4 + rr] = vv[e];
            }
        }
        // pW1: 3x64 -> 64 cols x 4 rows pack; 4th row carries pb1 (bias via K-pad)
        for (int i = tid; i < 256; i += 128) {
            int col = i >> 2, r = i & 3;
            smem[PW1_OFF + i] = (r < 3) ? pW1[r * 64 + col] : pb1[col];
        }
        // bias packs: row 0 = bias, rows 1..3 = 0 (consumed by the bias k-step)
        const float* bsrc[7] = {qb1, qb2, kb1, kb2, vb1, vb2, pb2};
        UNROLL for (int w = 0; w < 7; ++w) {
            for (int i = tid; i < 256; i += 128) {
                int col = i >> 2, r = i & 3;
                smem[BPACK_OFF + w * 256 + i] = (r == 0) ? bsrc[w][col] : 0.0f;
            }
        }
    }
    __syncthreads();

    const int tilesPerB = NPTS / 16;
    const int tile = blockIdx.x * 4 + wave;
    const int b  = tile / tilesPerB;
    const int n0 = (tile % tilesPerB) * 16;

    float* stage = &smem[STAGE_OFF + wave * (16 * STAGE_STRIDE)];
    float* qtl   = &smem[QTILE_OFF + wave * 1024];

    // constant A-fragment for the bias k-step: K0=1, K1..K3=0
    v2f ab;
    ab.x = hi ? 0.0f : 1.0f;
    ab.y = 0.0f;

    v2f afx[16], afh[16];
    v8f cK[4], cV[4], cP[4], ct[4];

    // ---- Q projection for the 16 queries of this tile ----
    {
        const float* xb = x + ((size_t)b * NPTS + n0) * DIM;
        load_afrag_global(xb, afx, lo, hi);
        mm16(afx, &smem[W_OFF(0)], &smem[BPACK_OFF + 0 * 256], ct, true, lo, hi, ab);
        store_c_stage(stage, ct, lo, hi);
        __syncthreads();
        load_afrag_stage(stage, afh, lo, hi);
        mm16(afh, &smem[W_OFF(1)], &smem[BPACK_OFF + 1 * 256], ct, false, lo, hi, ab);
        UNROLL for (int nb = 0; nb < 4; ++nb)
            UNROLL for (int r = 0; r < 8; ++r)
                qtl[(r + 8 * hi) * 64 + nb * 16 + lo] = ct[nb][r];
        __syncthreads();
    }

    // ---- per-query neighbor blocks: 16 neighbors = one 16-row tile ----
    for (int j = 0; j < 16; ++j) {
        const int n = n0 + j;
        const float* xnb = xn + (((size_t)b * NPTS + n) * KNBR) * DIM;
        load_afrag_global(xnb, afx, lo, hi);
        if (j < 15) {
            const char* nxt = (const char*)(xnb + KNBR * DIM);
            UNROLL for (int q = 0; q < 4; ++q)
                __builtin_prefetch(nxt + (lane * 4 + q) * 128, 0, 1);
        }

        // K projection
        mm16(afx, &smem[W_OFF(2)], &smem[BPACK_OFF + 2 * 256], ct, true, lo, hi, ab);
        store_c_stage(stage, ct, lo, hi);
        __syncthreads();
        load_afrag_stage(stage, afh, lo, hi);
        mm16(afh, &smem[W_OFF(3)], &smem[BPACK_OFF + 3 * 256], cK, false, lo, hi, ab);

        // V projection
        mm16(afx, &smem[W_OFF(4)], &smem[BPACK_OFF + 4 * 256], ct, true, lo, hi, ab);
        store_c_stage(stage, ct, lo, hi);
        __syncthreads();
        load_afrag_stage(stage, afh, lo, hi);
        mm16(afh, &smem[W_OFF(5)], &smem[BPACK_OFF + 5 * 256], cV, false, lo, hi, ab);

        // positional MLP layer 1: (16x3 pos-delta) @ (3x64) + pb1 as ONE WMMA:
        // A = (d0,d1)/(d2,1), B rows = pW1[0..2], pb1; C = inline zero.
        {
            const float* pp = p  + ((size_t)b * NPTS + n) * 3;
            const float* pq = pn + (((size_t)b * NPTS + n) * KNBR + lo) * 3;
            float d0 = pp[0] - pq[0];
            float d1 = pp[1] - pq[1];
            float d2 = pp[2] - pq[2];
            v2f a0;
            a0.x = hi ? d2 : d0;
            a0.y = hi ? 1.0f : d1;
            const float* w1 = &smem[PW1_OFF];
            const v8f zacc = {0.f, 0.f, 0.f, 0.f, 0.f, 0.f, 0.f, 0.f};
            UNROLL for (int nb = 0; nb < 4; ++nb) {
                v2f bf = *(const v2f*)(w1 + (nb * 16 + lo) * 4 + hi * 2);
                v8f acc = __builtin_amdgcn_wmma_f32_16x16x4_f32(
                    false, a0, false, bf, (short)0, zacc, false, false);
                UNROLL for (int r = 0; r < 8; ++r) acc[r] = fmaxf(acc[r], 0.0f);
                ct[nb] = acc;
            }
            store_c_stage(stage, ct, lo, hi);
            __syncthreads();
            load_afrag_stage(stage, afh, lo, hi);
        }
        // positional MLP layer 2 (64->64)
        mm16(afh, &smem[W_OFF(6)], &smem[BPACK_OFF + 6 * 256], cP, false, lo, hi, ab);

        // ---- softmax over the 16 neighbors (M axis of C layout) + weighted sum ----
        UNROLL for (int nb = 0; nb < 4; ++nb) {
            float qv = qtl[j * 64 + nb * 16 + lo];
            float lg[8];
            float m = -__builtin_inff();
            UNROLL for (int r = 0; r < 8; ++r) {
                lg[r] = qv - cK[nb][r] + cP[nb][r];
                m = fmaxf(m, lg[r]);
            }
            m = fmaxf(m, swap16(m));          // combine M=0..7 with M=8..15
            float s = 0.0f, o = 0.0f;
            UNROLL for (int r = 0; r < 8; ++r) {
                float e = __expf(lg[r] - m);
                s += e;
                o += e * (cV[nb][r] + cP[nb][r]);
            }
            s += swap16(s);
            o += swap16(o);
            float res = o / s;
            if (hi == 0)
                out[((size_t)b * NPTS + n) * DIM + nb * 16 + lo] = res;
        }
    }
}

extern "C" void kernel_launch(void* const* d_in, const int* in_sizes, int n_in,
                              void* d_out, int out_size, void* d_ws, size_t ws_size,
                              hipStream_t stream) {
    const float* x   = (const float*)d_in[0];
    const float* xn  = (const float*)d_in[1];
    const float* p   = (const float*)d_in[2];
    const float* pn  = (const float*)d_in[3];
    const float* qW1 = (const float*)d_in[4];
    const float* qb1 = (const float*)d_in[5];
    const float* qW2 = (const float*)d_in[6];
    const float* qb2 = (const float*)d_in[7];
    const float* kW1 = (const float*)d_in[8];
    const float* kb1 = (const float*)d_in[9];
    const float* kW2 = (const float*)d_in[10];
    const float* kb2 = (const float*)d_in[11];
    const float* vW1 = (const float*)d_in[12];
    const float* vb1 = (const float*)d_in[13];
    const float* vW2 = (const float*)d_in[14];
    const float* vb2 = (const float*)d_in[15];
    const float* pW1 = (const float*)d_in[16];
    const float* pb1 = (const float*)d_in[17];
    const float* pW2 = (const float*)d_in[18];
    const float* pb2 = (const float*)d_in[19];
    float* out = (float*)d_out;

    const size_t shmem = (size_t)SMEM_FLOATS * sizeof(float);  // ~153 KB (WGP allows 320 KB)
    hipFuncSetAttribute(reinterpret_cast<const void*>(&pt_fused_kernel),
                        hipFuncAttributeMaxDynamicSharedMemorySize, (int)shmem);

    const int totalTiles = (BATCH * NPTS) / 16;   // 2048
    dim3 grid(totalTiles / 4);                    // 4 waves (tiles) per workgroup
    dim3 block(128);
    hipLaunchKernelGGL(pt_fused_kernel, grid, block, shmem, stream,
                       x, xn, p, pn,
                       qW1, qb1, qW2, qb2,
                       kW1, kb1, kW2, kb2,
                       vW1, vb1, vW2, vb2,
                       pW1, pb1, pW2, pb2,
                       out);
}